// CustomAnyConv2d_36283883717257
// MI455X (gfx1250) — compile-verified
//
#include <hip/hip_runtime.h>
#include <stdint.h>

// Problem constants (fixed by the reference)
#define BATCH 8
#define CIN   64
#define COUT  128
#define HW    56
#define LTAPS 576   // 64*3*3

#define LUT_BYTES        (256 * 256 * 4)            // 262144 B, whole LUT in LDS
#define PATCH_ROW_COLS   64                         // padded row (iw -1..56 lives at c=3..60, 16B aligned interior at c=4)
#define PATCH_STRIDE_CI  (3 * PATCH_ROW_COLS)       // 192 dwords per cin
#define PATCH_DWORDS     (CIN * PATCH_STRIDE_CI)    // 12288 dwords
#define PATCH_BYTES      (PATCH_DWORDS * 4)         // 49152 B
#define SMEM_BYTES       (LUT_BYTES + PATCH_BYTES)  // 311296 B  (<= 320KB WGP LDS)

// ---- CDNA5 async global->LDS copy (ASYNCcnt path) -------------------------
__device__ __forceinline__ void async_load_b128(uint32_t lds_addr, const void* gptr) {
#if defined(__gfx1250__)
  asm volatile("global_load_async_to_lds_b128 %0, %1, off"
               :: "v"(lds_addr), "v"((uint64_t)(uintptr_t)gptr)
               : "memory");
#endif
}
__device__ __forceinline__ void wait_async_zero() {
#if defined(__gfx1250__)
  asm volatile("s_wait_asynccnt 0" ::: "memory");
#endif
}

// ---- Kernel A: transpose weight codes -> Wt[l][co] as u8 (72KB) -----------
__global__ void pack_weights_kernel(const int* __restrict__ weight,
                                    unsigned char* __restrict__ wt) {
  int t = blockIdx.x * 256 + threadIdx.x;
  if (t < LTAPS * COUT) {
    int l  = t >> 7;        // tap
    int co = t & 127;       // channel
    wt[t] = (unsigned char)(weight[co * LTAPS + l] & 0xFF);  // wt[l*128+co]
  }
}

// ---- Kernel B: LUT-in-LDS gather conv ------------------------------------
__global__ __launch_bounds__(256)
void lutconv_main_kernel(const int* __restrict__ input,
                         const unsigned* __restrict__ wt32,   // Wt as u32 (4 codes/lane)
                         const float* __restrict__ lut,
                         const float* __restrict__ bias,
                         float* __restrict__ out) {
  extern __shared__ char smem[];
  float* lut_s = (float*)smem;
  int*   patch = (int*)(smem + LUT_BYTES);
  float* outT  = (float*)(smem + LUT_BYTES);   // reused after compute

  const int t    = threadIdx.x;
  const int lane = t & 31;
  const int wave = t >> 5;
  const int blk  = blockIdx.x;
  const int b    = blk / HW;
  const int oh   = blk - b * HW;

  // 1) Stage the full 256KB LUT via async copies: 16384 x b128 chunks,
  //    lane-contiguous so each wave instruction streams 512B.
  {
    const char* gl = (const char*)lut;
    #pragma unroll 4
    for (int i = 0; i < 64; ++i) {
      int chunk = i * 256 + t;
      async_load_b128((uint32_t)(uintptr_t)(smem + chunk * 16), gl + chunk * 16);
    }
  }

  // 2) Zero the halo tile with DS stores while the LUT streams in.
  for (int i = t; i < PATCH_DWORDS; i += 256) patch[i] = 0;
  __syncthreads();  // zeros complete (DScnt) before async writes may overlap

  // 3) Stage valid input rows (ih = oh-1..oh+1) as async b128 copies.
  //    Interior iw=0..55 lands at c=4..59 (16B-aligned); borders stay code 0,
  //    matching jnp.pad zero-codes (which contribute lut[0, w]).
  {
    const char* gin = (const char*)(input + (size_t)b * CIN * HW * HW);
    char* pbase = (char*)patch;
    for (int c = t; c < 192 * 14; c += 256) {     // 192 runs x 14 b128 chunks
      int run = c / 14, k = c - run * 14;
      int ci  = run / 3, r = run - ci * 3;
      int ih  = oh - 1 + r;
      if ((unsigned)ih < (unsigned)HW) {
        const char* g = gin + ((size_t)(ci * HW + ih) * HW) * 4 + k * 16;
        uint32_t lo = (uint32_t)(uintptr_t)
            (pbase + ((ci * PATCH_STRIDE_CI + r * PATCH_ROW_COLS + 4) * 4) + k * 16);
        async_load_b128(lo, g);
      }
    }
  }
  wait_async_zero();
  __syncthreads();

  // 4) Gather-accumulate. Lane owns co = lane*4..lane*4+3; wave owns 7 pixels.
  float acc[7][4];
  #pragma unroll
  for (int p = 0; p < 7; ++p)
    acc[p][0] = acc[p][1] = acc[p][2] = acc[p][3] = 0.0f;

  const int ow0 = wave * 7;

  for (int ci = 0; ci < CIN; ++ci) {
    const int* prowb = patch + ci * PATCH_STRIDE_CI;
    #pragma unroll
    for (int kh = 0; kh < 3; ++kh) {
      const int* prow = prowb + kh * PATCH_ROW_COLS;
      #pragma unroll
      for (int kw = 0; kw < 3; ++kw) {
        const int l = ci * 9 + kh * 3 + kw;
        if ((l & 15) == 0 && l + 16 < LTAPS)
          __builtin_prefetch(&wt32[(l + 16) * 32 + lane], 0, 0);  // global_prefetch
        const unsigned wc = wt32[l * 32 + lane];
        const unsigned w0 = wc & 255u, w1 = (wc >> 8) & 255u;
        const unsigned w2 = (wc >> 16) & 255u, w3 = wc >> 24;
        #pragma unroll
        for (int p = 0; p < 7; ++p) {
          const int a = prow[ow0 + p + kw + 3];        // LDS broadcast read
          const float* row = lut_s + ((unsigned)a << 8);
          acc[p][0] += row[w0];
          acc[p][1] += row[w1];
          acc[p][2] += row[w2];
          acc[p][3] += row[w3];
        }
      }
    }
  }

  // 5) Bias + LDS transpose so global stores are contiguous along ow.
  __syncthreads();                       // everyone done reading patch
  const float4 bb = ((const float4*)bias)[lane];
  #pragma unroll
  for (int p = 0; p < 7; ++p) {
    const int ow = ow0 + p;
    outT[(lane * 4 + 0) * HW + ow] = acc[p][0] + bb.x;
    outT[(lane * 4 + 1) * HW + ow] = acc[p][1] + bb.y;
    outT[(lane * 4 + 2) * HW + ow] = acc[p][2] + bb.z;
    outT[(lane * 4 + 3) * HW + ow] = acc[p][3] + bb.w;
  }
  __syncthreads();

  float* ob = out + (size_t)b * COUT * HW * HW + (size_t)oh * HW;
  for (int idx = t; idx < COUT * HW; idx += 256) {
    const int co = idx / HW;
    const int ow = idx - co * HW;
    ob[(size_t)co * HW * HW + ow] = outT[idx];
  }
}

// ---- Host launcher --------------------------------------------------------
extern "C" void kernel_launch(void* const* d_in, const int* in_sizes, int n_in,
                              void* d_out, int out_size, void* d_ws, size_t ws_size,
                              hipStream_t stream) {
  const int*   input  = (const int*)d_in[0];    // (8,64,56,56) int32 codes
  const int*   weight = (const int*)d_in[1];    // (128,64,3,3) int32 codes
  const float* lut    = (const float*)d_in[2];  // (256,256) f32
  const float* bias   = (const float*)d_in[3];  // (128,) f32
  float*       out    = (float*)d_out;          // (8,128,56,56) f32
  unsigned char* wt   = (unsigned char*)d_ws;   // 72KB packed weight codes

  hipFuncSetAttribute((const void*)lutconv_main_kernel,
                      hipFuncAttributeMaxDynamicSharedMemorySize, SMEM_BYTES);

  pack_weights_kernel<<<(LTAPS * COUT + 255) / 256, 256, 0, stream>>>(weight, wt);
  lutconv_main_kernel<<<BATCH * HW, 256, SMEM_BYTES, stream>>>(
      input, (const unsigned*)d_ws, lut, bias, out);
}